// QGate_65481071403203
// MI455X (gfx1250) — compile-verified
//
#include <hip/hip_runtime.h>
#include <hip/hip_bf16.h>

typedef float v2f __attribute__((ext_vector_type(2)));
typedef float v8f __attribute__((ext_vector_type(8)));

#define NQ       14
#define NSTATES  (1 << NQ)        // 16384
#define NPAIRS   (1 << (NQ - 1))  // 8192
#define BLOCK    256
#define NWAVES   (BLOCK / 32)     // 8
#define GSTRIDE  (NWAVES * 16)    // 128 pairs per workgroup step
#define GITER    (NPAIRS / GSTRIDE) // 64 WMMA iterations per wave per gate

// One gate micro-step: D = A(16x4 gate) x B(4x16 = this lane's float2), take D[0],D[1].
__device__ __forceinline__ float2 gate_mm(v2f A, float2 v) {
  v2f B; B.x = v.x; B.y = v.y;
  v8f C = {};
  v8f D = __builtin_amdgcn_wmma_f32_16x16x4_f32(
      false, A, false, B, (short)0, C, false, false);
  float2 o; o.x = D[0]; o.y = D[1];
  return o;
}

__device__ __forceinline__ int gate_idx(int pair0, int it, int p, int off) {
  const int pn = pair0 + it * GSTRIDE;
  return pn + ((pn >> p) << p) + off;   // insert 0-bit at position p, add hh*stride
}

// Apply 1-qubit complex 2x2 gate U=[[a00+ib00,a01+ib01],[a10+ib10,a11+ib11]] to the
// qubit at flat-index bit position p (pair stride 2^p). State in LDS, interleaved
// float2 (re,im).
//
// Real 4x4 rep G acting on (r0,i0,r1,i1):
//   row0: [ ar00 -ai00  ar01 -ai01 ]   (new_r0)
//   row1: [ ai00  ar00  ai01  ar01 ]   (new_i0)
//   row2: [ ar10 -ai10  ar11 -ai11 ]   (new_r1)
//   row3: [ ai10  ar10  ai11  ar11 ]   (new_i1)
//
// WMMA f32 16x16x4: A rows 0..3 = G rows 0..3, A rows 8..9 = G rows 2..3 (duplicate).
// D lanes 0-15 hold rows 0..7, lanes 16-31 rows 8..15, so every lane's D[0],D[1] are
// exactly the new (re,im) of the float2 it loaded at i0 + hh*st. Fully convergent:
// 1x ds_load_b64 + 1x v_wmma + 1x ds_store_b64 per lane per 16 pairs/wave.
//
// 3-stage software pipeline per iteration:
//   load(it+1) ; store(it-1) <- result of previous WMMA ; wmma(it)
// The deferred store breaks the WAR between the in-flight store (reading the old
// WMMA result VGPRs) and the next WMMA (writing them): with unroll 2 the two result
// registers alternate, so DScnt never needs a full drain inside the loop. Addresses
// across stages are disjoint; LDS is in-order per wave; barrier follows all stores.
__device__ __forceinline__ void apply_gate_wmma(
    float2* st, int p,
    float ar00, float ai00, float ar01, float ai01,
    float ar10, float ai10, float ar11, float ai11)
{
  const int tid  = threadIdx.x;
  const int lane = tid & 31;
  const int wave = tid >> 5;
  const int m    = lane & 15;
  const int hh   = lane >> 4;     // K columns: 0 -> {0,1}, 1 -> {2,3}
  // A element (ISA 16x4 f32 layout: VGPR0 = K(2hh), VGPR1 = K(2hh+1), lane%16 = M row)
  const bool valid = (m < 4) || (m == 8) || (m == 9);
  const int  r     = (m >= 4) ? 1 : (m >> 1);
  const int  comp  = m & 1;
  const float ur = r ? (hh ? ar11 : ar10) : (hh ? ar01 : ar00);
  const float ui = r ? (hh ? ai11 : ai10) : (hh ? ai01 : ai00);
  v2f A;
  A.x = valid ? (comp ? ui :  ur) : 0.0f;
  A.y = valid ? (comp ? ur : -ui) : 0.0f;

  const int stq   = 1 << p;
  const int off   = hh ? stq : 0;
  const int pair0 = wave * 16 + m;

  int    id_st = gate_idx(pair0, 0, p, off);   // pending store target (it-1)
  int    id_ld = gate_idx(pair0, 1, p, off);   // most recent load index (it)
  float2 v_cur = st[id_st];                    // v(0)
  float2 v_nxt = st[id_ld];                    // v(1)
  float2 o_prev = gate_mm(A, v_cur);           // o(0)

#pragma unroll 2
  for (int it = 1; it < GITER - 1; ++it) {
    const int    id2 = gate_idx(pair0, it + 1, p, off);
    const float2 v2  = st[id2];                // load(it+1)
    st[id_st] = o_prev;                        // store(it-1)
    o_prev    = gate_mm(A, v_nxt);             // wmma(it)
    id_st = id_ld; id_ld = id2;
    v_nxt = v2;
  }
  st[id_st] = o_prev;                          // store(GITER-2)
  st[id_ld] = gate_mm(A, v_nxt);               // wmma+store(GITER-1)
}

// CNOT: control bit pc, target bit pt; swap (c=1,t=0) <-> (c=1,t=1).
__device__ __forceinline__ void apply_cnot(float2* st, int pc, int pt)
{
  const int tid = threadIdx.x;
  const int lo  = pc < pt ? pc : pt;
  const int hi  = pc < pt ? pt : pc;
#pragma unroll 2
  for (int it = 0; it < (NSTATES >> 2) / BLOCK; ++it) {
    const int j = tid + it * BLOCK;
    int t  = j + ((j >> lo) << lo);   // insert 0-bit at lo
    int i  = t + ((t >> hi) << hi);   // insert 0-bit at hi
    i     |= (1 << pc);
    const int i2 = i | (1 << pt);
    const float2 a = st[i], b2 = st[i2];
    st[i] = b2; st[i2] = a;
  }
}

__global__ void __launch_bounds__(BLOCK)
qsim_kernel(const float* __restrict__ x, const float* __restrict__ params,
            float* __restrict__ out, int depth)
{
  __shared__ float2 st[NSTATES];   // 128 KB (2 WGs fit in the 320 KB WGP)
  __shared__ float  red[16];

  const int b   = blockIdx.x;
  const int tid = threadIdx.x;

  // |0...0>
#pragma unroll 4
  for (int it = 0; it < NSTATES / BLOCK; ++it)
    st[tid + it * BLOCK] = make_float2(0.0f, 0.0f);
  if (tid == 0)  st[0]    = make_float2(1.0f, 0.0f);
  if (tid < 16)  red[tid] = 0.0f;
  __syncthreads();

  // Data-encoding RX(x[b,q]); wire q <-> flat-index bit (13-q).
  for (int q = 0; q < NQ; ++q) {
    const float th = x[b * NQ + q];
    float s, c;
    __sincosf(0.5f * th, &s, &c);
    // RX: u00=c, u01=-i s, u10=-i s, u11=c
    apply_gate_wmma(st, (NQ - 1) - q, c, 0.0f, 0.0f, -s, 0.0f, -s, c, 0.0f);
    __syncthreads();
  }

  for (int l = 0; l < depth; ++l) {
    for (int q = 0; q < NQ; ++q) {
      const float t0 = params[(l * NQ + q) * 3 + 0];
      const float t1 = params[(l * NQ + q) * 3 + 1];
      const float t2 = params[(l * NQ + q) * 3 + 2];
      float s0, c0, s1, c1, sz, cz;
      __sincosf(0.5f * t0, &s0, &c0);
      __sincosf(0.5f * t1, &s1, &c1);
      __sincosf(0.5f * t2, &sz, &cz);
      // Fused U = RZ(t2) * RY(t1) * RX(t0)
      const float ar00 =  cz*c1*c0 + sz*s1*s0;
      const float ai00 =  cz*s1*s0 - sz*c1*c0;
      const float ar01 = -cz*s1*c0 - sz*c1*s0;
      const float ai01 =  sz*s1*c0 - cz*c1*s0;
      const float ar10 =  cz*s1*c0 + sz*c1*s0;
      const float ai10 =  sz*s1*c0 - cz*c1*s0;
      const float ar11 =  cz*c1*c0 + sz*s1*s0;
      const float ai11 =  sz*c1*c0 - cz*s1*s0;
      apply_gate_wmma(st, (NQ - 1) - q,
                      ar00, ai00, ar01, ai01, ar10, ai10, ar11, ai11);
      __syncthreads();
    }
    // CNOT ring: (q -> q+1) for q=0..12, then (13 -> 0)
    for (int q = 0; q < NQ - 1; ++q) {
      apply_cnot(st, (NQ - 1) - q, (NQ - 2) - q);
      __syncthreads();
    }
    apply_cnot(st, 0, NQ - 1);
    __syncthreads();
  }

  // <Z_w> = sum_s |amp_s|^2 * (1 - 2*bit_{13-w}(s))
  float acc[NQ];
#pragma unroll
  for (int w = 0; w < NQ; ++w) acc[w] = 0.0f;
#pragma unroll 2
  for (int it = 0; it < NSTATES / BLOCK; ++it) {
    const int i = tid + it * BLOCK;
    const float2 a = st[i];
    const float pr = a.x * a.x + a.y * a.y;
#pragma unroll
    for (int w = 0; w < NQ; ++w)
      acc[w] += ((i >> ((NQ - 1) - w)) & 1) ? -pr : pr;
  }
#pragma unroll
  for (int w = 0; w < NQ; ++w) atomicAdd(&red[w], acc[w]);   // ds_add_f32
  __syncthreads();
  if (tid < NQ) out[b * NQ + tid] = red[tid];
}

extern "C" void kernel_launch(void* const* d_in, const int* in_sizes, int n_in,
                              void* d_out, int out_size, void* d_ws, size_t ws_size,
                              hipStream_t stream) {
  const float* x      = (const float*)d_in[0];   // (batch, 14) float32
  const float* params = (const float*)d_in[1];   // (depth, 14, 3) float32
  float*       out    = (float*)d_out;           // (batch, 14) float32
  const int batch = in_sizes[0] / NQ;
  const int depth = in_sizes[1] / (NQ * 3);
  hipLaunchKernelGGL(qsim_kernel, dim3(batch), dim3(BLOCK), 0, stream,
                     x, params, out, depth);
}